// dense_res_bit_tree_meanvar_baens_61942018343214
// MI455X (gfx1250) — compile-verified
//
#include <hip/hip_runtime.h>
#include <math.h>
#include <float.h>

// ---------------- problem constants (match reference) ----------------
#define NROWS   8
#define DIM1    1024
#define DIM2    1024
#define DCOLS   (DIM1*DIM2)          // 1,048,576
#define BATCH   1024
#define NDTOT   (NROWS*DCOLS)        // 8,388,608
#define PA_BLOCKS 1024               // pass-A / minmax reduction blocks

typedef __attribute__((ext_vector_type(2))) float v2f;
typedef __attribute__((ext_vector_type(4))) float v4f;
typedef __attribute__((ext_vector_type(8))) float v8f;

// scalars block layout (floats): [0]=beta [1]=alpha [2]=s0
// [8..15]=rowcoef[n] = grad_g[gid[n]]/max(counts[gid[n]],1)
// [16..23]=gid[n] (as float)

// ---------------- 1) global min/max of U ----------------
__global__ void kminmax(const float* __restrict__ U, float* __restrict__ part) {
    __shared__ float smax[256], smin[256];
    int tid = threadIdx.x;
    float mx = -FLT_MAX, mn = FLT_MAX;
    for (int i = blockIdx.x * 256 + tid; i < NDTOT; i += gridDim.x * 256) {
        float u = U[i];
        mx = fmaxf(mx, u); mn = fminf(mn, u);
    }
    smax[tid] = mx; smin[tid] = mn;
    __syncthreads();
    for (int s = 128; s > 0; s >>= 1) {
        if (tid < s) {
            smax[tid] = fmaxf(smax[tid], smax[tid + s]);
            smin[tid] = fminf(smin[tid], smin[tid + s]);
        }
        __syncthreads();
    }
    if (tid == 0) { part[blockIdx.x * 2 + 0] = smax[0]; part[blockIdx.x * 2 + 1] = smin[0]; }
}

__global__ void kminmax_final(const float* __restrict__ part, float* __restrict__ scal) {
    __shared__ float smax[256], smin[256];
    int tid = threadIdx.x;
    float mx = -FLT_MAX, mn = FLT_MAX;
    for (int b = tid; b < PA_BLOCKS; b += 256) {
        mx = fmaxf(mx, part[b * 2 + 0]);
        mn = fminf(mn, part[b * 2 + 1]);
    }
    smax[tid] = mx; smin[tid] = mn;
    __syncthreads();
    for (int s = 128; s > 0; s >>= 1) {
        if (tid < s) {
            smax[tid] = fmaxf(smax[tid], smax[tid + s]);
            smin[tid] = fminf(smin[tid], smin[tid + s]);
        }
        __syncthreads();
    }
    if (tid == 0) {
        float beta = smax[0], alpha = smin[0];
        scal[0] = beta; scal[1] = alpha;
        scal[2] = (beta - alpha) / 3.0f;   // RES_DENOS[0]
    }
}

// ---------------- 2) level 0: w = s0 * floor(U / s0) ----------------
__global__ void klevel0(const float* __restrict__ U, float* __restrict__ w,
                        const float* __restrict__ scal) {
    float s = scal[2];
    for (int i = blockIdx.x * 256 + threadIdx.x; i < NDTOT; i += gridDim.x * 256)
        w[i] = s * floorf(U[i] / s);
}

// ---------------- 3) pass A: per-column sort + delta stats ----------------
__global__ void kpassA(const float* __restrict__ U, const float* __restrict__ w,
                       float* __restrict__ part) {
    __shared__ float red[8 * 256];
    int tid = threadIdx.x;
    float acc[8];
#pragma unroll
    for (int q = 0; q < 8; ++q) acc[q] = 0.0f;   // deltas are >= 0, so max init 0 is safe

    for (int d = blockIdx.x * 256 + tid; d < DCOLS; d += gridDim.x * 256) {
        float v[NROWS];
#pragma unroll
        for (int n = 0; n < NROWS; ++n) v[n] = U[n * DCOLS + d] - w[n * DCOLS + d];
        // stable insertion sort ascending (values only)
#pragma unroll
        for (int i = 1; i < NROWS; ++i) {
            float key = v[i]; int j = i - 1;
            while (j >= 0 && v[j] > key) { v[j + 1] = v[j]; --j; }
            v[j + 1] = key;
        }
#pragma unroll
        for (int i = 0; i < NROWS - 1; ++i) {
            float dl = v[i + 1] - v[i];
            acc[i] += dl;
            acc[7] = fmaxf(acc[7], dl);
        }
    }
#pragma unroll
    for (int q = 0; q < 8; ++q) red[q * 256 + tid] = acc[q];
    __syncthreads();
    for (int s = 128; s > 0; s >>= 1) {
        if (tid < s) {
#pragma unroll
            for (int q = 0; q < 7; ++q) red[q * 256 + tid] += red[q * 256 + tid + s];
            red[7 * 256 + tid] = fmaxf(red[7 * 256 + tid], red[7 * 256 + tid + s]);
        }
        __syncthreads();
    }
    if (tid == 0) {
#pragma unroll
        for (int q = 0; q < 8; ++q) part[blockIdx.x * 8 + q] = red[q * 256];
    }
}

// ---------------- 4) finalize level stats -> rowcoef/gid ----------------
__global__ void kmid(const float* __restrict__ part, const float* __restrict__ thres_mean,
                     float* __restrict__ scal) {
    __shared__ float red[8 * 256];
    int tid = threadIdx.x;
    float acc[8];
#pragma unroll
    for (int q = 0; q < 8; ++q) acc[q] = 0.0f;
    for (int b = tid; b < PA_BLOCKS; b += 256) {
#pragma unroll
        for (int q = 0; q < 7; ++q) acc[q] += part[b * 8 + q];
        acc[7] = fmaxf(acc[7], part[b * 8 + 7]);
    }
#pragma unroll
    for (int q = 0; q < 8; ++q) red[q * 256 + tid] = acc[q];
    __syncthreads();
    for (int s = 128; s > 0; s >>= 1) {
        if (tid < s) {
#pragma unroll
            for (int q = 0; q < 7; ++q) red[q * 256 + tid] += red[q * 256 + tid + s];
            red[7 * 256 + tid] = fmaxf(red[7 * 256 + tid], red[7 * 256 + tid + s]);
        }
        __syncthreads();
    }
    if (tid == 0) {
        float dmax = red[7 * 256];
        float invDen = 1.0f / ((float)DCOLS * dmax);
        float thres = 1.0f / (1.0f + expf(-thres_mean[0]));
        float sp[7], bb[7];
#pragma unroll
        for (int i = 0; i < 7; ++i) {
            float mean = red[i * 256] * invDen;              // mean of normalized delta
            sp[i] = 1.0f / (1.0f + expf(-(mean - thres) / 0.01f));
            bb[i] = rintf(sp[i]);                            // round-half-even like jnp.round
        }
        int gid[8]; gid[0] = 0;
#pragma unroll
        for (int n = 1; n < 8; ++n) gid[n] = gid[n - 1] + (int)bb[n - 1];
        float counts[8], logsum[8];
#pragma unroll
        for (int g = 0; g < 8; ++g) { counts[g] = 0.0f; logsum[g] = 0.0f; }
#pragma unroll
        for (int n = 0; n < 8; ++n) counts[gid[n]] += 1.0f;
#pragma unroll
        for (int i = 0; i < 7; ++i) {
            float f = (bb[i] == 1.0f) ? sp[i] : (1.0f - sp[i]);
            logsum[gid[i]] += logf(f);
        }
#pragma unroll
        for (int n = 0; n < 8; ++n) {
            int g = gid[n];
            scal[8 + n] = expf(logsum[g]) / fmaxf(counts[g], 1.0f);
            scal[16 + n] = (float)g;
        }
    }
}

// ---------------- 5) pass B: grouped means, scatter, residual update ----------------
__global__ void kpassB(const float* __restrict__ U, float* __restrict__ w,
                       const float* __restrict__ scal, float s_factor) {
    int d = blockIdx.x * 256 + threadIdx.x;
    if (d >= DCOLS) return;
    float s = scal[2] * s_factor;
    int gid[8]; float rc[8];
#pragma unroll
    for (int n = 0; n < 8; ++n) { gid[n] = (int)scal[16 + n]; rc[n] = scal[8 + n]; }

    float v[NROWS]; int ord[NROWS];
#pragma unroll
    for (int n = 0; n < NROWS; ++n) { v[n] = U[n * DCOLS + d] - w[n * DCOLS + d]; ord[n] = n; }
    // stable insertion sort ascending, tracking the permutation (== jnp.argsort stable)
#pragma unroll
    for (int i = 1; i < NROWS; ++i) {
        float key = v[i]; int ki = ord[i]; int j = i - 1;
        while (j >= 0 && v[j] > key) { v[j + 1] = v[j]; ord[j + 1] = ord[j]; --j; }
        v[j + 1] = key; ord[j + 1] = ki;
    }
    float gsum[8];
#pragma unroll
    for (int g = 0; g < 8; ++g) gsum[g] = 0.0f;
#pragma unroll
    for (int k = 0; k < 8; ++k) gsum[gid[k]] += v[k];
    float grouped[8];
#pragma unroll
    for (int k = 0; k < 8; ++k) grouped[ord[k]] = rc[k] * gsum[gid[k]];
#pragma unroll
    for (int n = 0; n < 8; ++n)
        w[n * DCOLS + d] += s * floorf(grouped[n] / s);
}

// ---------------- 6) batched GEMM via V_WMMA_F32_16X16X4_F32 ----------------
// out[b, n, l] = sum_d1 x[b, n, d1] * w[n, d1, l]
// blockIdx.x: N-tile (128 cols), blockIdx.y: M-tile (128 rows of b), blockIdx.z: n
#define BK 32
#define ASTRIDE 36    // A tile [128][32] padded: conflict-free b64 frag reads, 16B-aligned stores
#define BPAIR   288   // B tile k-pair stride: 288 % 64 == 32 -> lane halves hit disjoint banks
__global__ void __launch_bounds__(256)
kgemm(const float* __restrict__ x, const float* __restrict__ w, float* __restrict__ out) {
    __shared__ float As[128 * ASTRIDE];         // As[m][k]
    __shared__ float Bs[(BK / 2) * BPAIR];      // Bs[k/2][n*2 + (k&1)]  (k-pair interleaved)

    const int tid  = threadIdx.x;
    const int lane = tid & 31;
    const int wave = tid >> 5;           // 8 waves
    const int wm   = wave & 3;           // 4 waves over M (32 rows each)
    const int wn   = wave >> 2;          // 2 waves over N (64 cols each)
    const int nz   = blockIdx.z;
    const int m0   = blockIdx.y * 128;
    const int n0   = blockIdx.x * 128;

    const int lhalf = lane >> 4;         // 0: lanes 0-15, 1: lanes 16-31
    const int l16   = lane & 15;

    v8f acc[2][4];
    const v8f zero = {};
#pragma unroll
    for (int i = 0; i < 2; ++i)
#pragma unroll
        for (int j = 0; j < 4; ++j) acc[i][j] = zero;

    // staging indices (constant per thread)
    const int aRow[4] = { (tid + 0*256) >> 3, (tid + 1*256) >> 3,
                          (tid + 2*256) >> 3, (tid + 3*256) >> 3 };
    const int aCol = (tid & 7) * 4;
    const int bKp[2]  = { tid >> 5, (tid >> 5) + 8 };   // k-pair index per r
    const int bNq  = (tid & 31) * 4;                    // n quad base

    v4f ra[4], rb[4], ra2[4], rb2[4];

    // prologue: fetch first K tile into registers
#pragma unroll
    for (int r = 0; r < 4; ++r)
        ra[r] = *(const v4f*)&x[((size_t)(m0 + aRow[r]) * NROWS + nz) * DIM1 + aCol];
#pragma unroll
    for (int r = 0; r < 2; ++r) {
        int k = 2 * bKp[r];
        rb[2 * r + 0] = *(const v4f*)&w[((size_t)nz * DIM1 + k + 0) * DIM2 + n0 + bNq];
        rb[2 * r + 1] = *(const v4f*)&w[((size_t)nz * DIM1 + k + 1) * DIM2 + n0 + bNq];
    }

    for (int kb = 0; kb < DIM1; kb += BK) {
        // stage current tile regs -> LDS
#pragma unroll
        for (int r = 0; r < 4; ++r)
            *(v4f*)&As[aRow[r] * ASTRIDE + aCol] = ra[r];
#pragma unroll
        for (int r = 0; r < 2; ++r)
#pragma unroll
            for (int c = 0; c < 4; ++c) {
                v2f p; p.x = rb[2 * r + 0][c]; p.y = rb[2 * r + 1][c];
                *(v2f*)&Bs[bKp[r] * BPAIR + (bNq + c) * 2] = p;
            }
        __syncthreads();

        // prefetch next tile into registers (overlaps with WMMA below)
        const bool hasNext = (kb + BK) < DIM1;
        if (hasNext) {
            const int kn = kb + BK;
#pragma unroll
            for (int r = 0; r < 4; ++r)
                ra2[r] = *(const v4f*)&x[((size_t)(m0 + aRow[r]) * NROWS + nz) * DIM1 + kn + aCol];
#pragma unroll
            for (int r = 0; r < 2; ++r) {
                int k = kn + 2 * bKp[r];
                rb2[2 * r + 0] = *(const v4f*)&w[((size_t)nz * DIM1 + k + 0) * DIM2 + n0 + bNq];
                rb2[2 * r + 1] = *(const v4f*)&w[((size_t)nz * DIM1 + k + 1) * DIM2 + n0 + bNq];
            }
        }

#pragma unroll
        for (int kk = 0; kk < BK / 4; ++kk) {
            const int k0 = kk * 4;
            // A fragments (16x4 f32): ISA 7.12.2 — VGPR0={K0|K2}, VGPR1={K1|K3}
            v2f afrag[2];
#pragma unroll
            for (int i = 0; i < 2; ++i) {
                int m = wm * 32 + i * 16 + l16;
                afrag[i] = *(const v2f*)&As[m * ASTRIDE + k0 + lhalf * 2];
            }
            // B fragments (4x16 f32): pair-interleaved layout -> one b64 per fragment
            v2f bfrag[4];
#pragma unroll
            for (int j = 0; j < 4; ++j) {
                int n = wn * 64 + j * 16 + l16;
                bfrag[j] = *(const v2f*)&Bs[(2 * kk + lhalf) * BPAIR + n * 2];
            }
#pragma unroll
            for (int i = 0; i < 2; ++i)
#pragma unroll
                for (int j = 0; j < 4; ++j)
                    acc[i][j] = __builtin_amdgcn_wmma_f32_16x16x4_f32(
                        false, afrag[i], false, bfrag[j],
                        (short)0, acc[i][j], false, false);
        }
        __syncthreads();

        if (hasNext) {
#pragma unroll
            for (int r = 0; r < 4; ++r) { ra[r] = ra2[r]; rb[r] = rb2[r]; }
        }
    }

    // epilogue: C/D layout — VGPR v holds rows v (lanes 0-15) and v+8 (lanes 16-31)
#pragma unroll
    for (int i = 0; i < 2; ++i)
#pragma unroll
        for (int j = 0; j < 4; ++j) {
            int col = n0 + wn * 64 + j * 16 + l16;
#pragma unroll
            for (int v = 0; v < 8; ++v) {
                int row = m0 + wm * 32 + i * 16 + v + 8 * lhalf;
                out[((size_t)row * NROWS + nz) * DIM2 + col] = acc[i][j][v];
            }
        }
}

// ---------------- launcher ----------------
extern "C" void kernel_launch(void* const* d_in, const int* in_sizes, int n_in,
                              void* d_out, int out_size, void* d_ws, size_t ws_size,
                              hipStream_t stream) {
    const float* x  = (const float*)d_in[0];          // [B, N, D1]
    const float* U  = (const float*)d_in[1];          // [N, D]
    const float* tm = (const float*)d_in[2];          // [1]
    float* out = (float*)d_out;                       // [B, N, D2]

    float* w    = (float*)d_ws;                       // NDTOT floats (quantized weights)
    float* part = w + NDTOT;                          // PA_BLOCKS*8 floats
    float* scal = part + PA_BLOCKS * 8;               // 32 floats

    // s0 = (max-min)/3
    kminmax<<<dim3(PA_BLOCKS), dim3(256), 0, stream>>>(U, part);
    kminmax_final<<<dim3(1), dim3(256), 0, stream>>>(part, scal);
    // level 0
    klevel0<<<dim3(8192), dim3(256), 0, stream>>>(U, w, scal);
    // level 1 (s = s0/5) and level 2 (s = s0/85)
    const float sfac[2] = {1.0f / 5.0f, 1.0f / 85.0f};
    for (int lvl = 0; lvl < 2; ++lvl) {
        kpassA<<<dim3(PA_BLOCKS), dim3(256), 0, stream>>>(U, w, part);
        kmid<<<dim3(1), dim3(256), 0, stream>>>(part, tm, scal);
        kpassB<<<dim3(DCOLS / 256), dim3(256), 0, stream>>>(U, w, scal, sfac[lvl]);
    }
    // einsum('bnd,ndl->bnl'): 8 GEMMs of 1024x1024x1024, fp32 WMMA
    kgemm<<<dim3(DIM2 / 128, BATCH / 128, NROWS), dim3(256), 0, stream>>>(x, w, out);
}